// GNNPolicy_82678120448124
// MI455X (gfx1250) — compile-verified
//
#include <hip/hip_runtime.h>

#define D 32

typedef float v2f __attribute__((ext_vector_type(2)));
typedef float v8f __attribute__((ext_vector_type(8)));

// ---------------- degree / normalization ----------------

__global__ void k_deg_init(float* __restrict__ deg, int N) {
  int i = blockIdx.x * blockDim.x + threadIdx.x;
  if (i < N) deg[i] = 1.0f;   // self-loop
}

__global__ void k_deg_edges(const int* __restrict__ dst, float* __restrict__ deg, int E) {
  int e = blockIdx.x * blockDim.x + threadIdx.x;
  if (e < E) atomicAdd(&deg[dst[e]], 1.0f);
}

__global__ void k_rsqrt_inplace(float* __restrict__ deg, int N) {
  int i = blockIdx.x * blockDim.x + threadIdx.x;
  if (i < N) deg[i] = rsqrtf(deg[i]);
}

// ---------------- fp32 WMMA GEMM: Out[N,32] = op(A[N,32]) @ W[32,32] ----------------
// One wave computes a 16-row x 32-col output tile via two 16x16 accumulators,
// K=32 as 8 chained V_WMMA_F32_16X16X4_F32 per accumulator.
__global__ void k_gemm32(const float* __restrict__ A, const float* __restrict__ W,
                         float* __restrict__ Out, int N, int relu) {
  int wave = blockIdx.x * (blockDim.x >> 5) + (threadIdx.x >> 5);
  int lane = threadIdx.x & 31;
  int row0 = wave << 4;
  if (row0 >= N) return;                 // wave-uniform exit, EXEC stays all-ones

  int m  = lane & 15;                    // A row within tile / B-N column
  int kh = lane >> 4;                    // K half (ISA 16x4 fp32 A layout)
  int row = row0 + m; if (row >= N) row = N - 1;   // clamp keeps EXEC full
  const float2* ap = (const float2*)(A + (size_t)row * D);

  v8f c0 = {}; v8f c1 = {};
#pragma unroll
  for (int j = 0; j < 8; ++j) {
    float2 t = ap[2 * j + kh];           // A[m][4j+2kh], A[m][4j+2kh+1]
    v2f a; a.x = t.x; a.y = t.y;
    if (relu) { a.x = fmaxf(a.x, 0.f); a.y = fmaxf(a.y, 0.f); }
    int k0 = 4 * j + 2 * kh;
    v2f b0, b1;                          // B rows striped across lanes (N = m)
    b0.x = W[k0 * D + m];        b0.y = W[(k0 + 1) * D + m];
    b1.x = W[k0 * D + 16 + m];   b1.y = W[(k0 + 1) * D + 16 + m];
    c0 = __builtin_amdgcn_wmma_f32_16x16x4_f32(false, a, false, b0, (short)0, c0, false, false);
    c1 = __builtin_amdgcn_wmma_f32_16x16x4_f32(false, a, false, b1, (short)0, c1, false, false);
  }

  int n  = lane & 15;
  int mb = (lane >> 4) << 3;             // C/D layout: M = v + 8*(lane>>4)
#pragma unroll
  for (int v = 0; v < 8; ++v) {
    int r = row0 + mb + v;
    if (r < N) {
      Out[(size_t)r * D + n]      = c0[v];
      Out[(size_t)r * D + 16 + n] = c1[v];
    }
  }
}

// ---------------- self-loop + bias init: out = t * dis^2 + b ----------------
__global__ void k_self_bias(const float* __restrict__ t, const float* __restrict__ dis,
                            const float* __restrict__ b, float* __restrict__ out, int N) {
  int i = blockIdx.x * blockDim.x + threadIdx.x;
  int node = i >> 5, d = i & 31;
  if (node >= N) return;
  float di = dis[node];
  out[(size_t)node * D + d] = t[(size_t)node * D + d] * (di * di) + b[d];
}

// ---------------- edge scatter-add: agg[dst] += t[src] * dis[src]*dis[dst] ----------------
// One wave per edge: 32 lanes = 32 features -> coalesced 128B gather, L2 atomics.
__global__ void k_edge_agg(const float* __restrict__ t, const float* __restrict__ dis,
                           const int* __restrict__ src, const int* __restrict__ dst,
                           float* __restrict__ agg, int E) {
  int idx = blockIdx.x * blockDim.x + threadIdx.x;
  int e = idx >> 5;
  int d = idx & 31;
  if (e >= E) return;
  int s = src[e], u = dst[e];
  float norm = dis[s] * dis[u];
  float val = t[(size_t)s * D + d] * norm;
  atomicAdd(&agg[(size_t)u * D + d], val);
}

// ---------------- final row dot-products ----------------
__global__ void k_dot(const float* __restrict__ y1, const float* __restrict__ y2,
                      const int* __restrict__ op1, const int* __restrict__ op2,
                      float* __restrict__ out) {
  const float* y = (blockIdx.x == 0) ? y1 : y2;
  int o1 = op1[0], o2 = op2[0];
  int d = threadIdx.x;                   // 32 threads = one wave
  float p = y[(size_t)o1 * D + d] * y[(size_t)o2 * D + d];
#pragma unroll
  for (int off = 16; off; off >>= 1) p += __shfl_down(p, off, 32);
  if (d == 0) out[blockIdx.x] = p;
}

extern "C" void kernel_launch(void* const* d_in, const int* in_sizes, int n_in,
                              void* d_out, int out_size, void* d_ws, size_t ws_size,
                              hipStream_t stream) {
  const float* x   = (const float*)d_in[0];
  const int*   ei  = (const int*)  d_in[1];
  const float* W1a = (const float*)d_in[2];  const float* b1a = (const float*)d_in[3];
  const float* W2a = (const float*)d_in[4];  const float* b2a = (const float*)d_in[5];
  const float* W1b = (const float*)d_in[6];  const float* b1b = (const float*)d_in[7];
  const float* W2b = (const float*)d_in[8];  const float* b2b = (const float*)d_in[9];
  const int*   op1 = (const int*)  d_in[10];
  const int*   op2 = (const int*)  d_in[11];

  const int N = in_sizes[0] / D;
  const int E = in_sizes[1] / 2;
  const int* src = ei;          // edge_index[0]
  const int* dst = ei + E;      // edge_index[1]

  // workspace layout (floats): dis[N] | t[N*D] | agg[N*D] | y1[N*D] | y2[N*D]
  float* ws  = (float*)d_ws;
  size_t npad = ((size_t)N + 255) & ~(size_t)255;
  float* dis = ws;
  float* t   = dis + npad;
  float* agg = t   + (size_t)N * D;
  float* y1  = agg + (size_t)N * D;
  float* y2  = y1  + (size_t)N * D;

  const int TPB = 256;
  int nBlocksN   = (N + TPB - 1) / TPB;
  int nBlocksE   = (E + TPB - 1) / TPB;
  int nBlocksND  = (int)(((size_t)N * D + TPB - 1) / TPB);
  int nBlocksED  = (int)(((size_t)E * D + TPB - 1) / TPB);
  int gemmBlocks = (((N + 15) / 16) + 7) / 8;   // 8 waves per block, 16 rows per wave

  // normalization: dis = rsqrt(deg + 1)
  k_deg_init<<<nBlocksN, TPB, 0, stream>>>(dis, N);
  k_deg_edges<<<nBlocksE, TPB, 0, stream>>>(dst, dis, E);
  k_rsqrt_inplace<<<nBlocksN, TPB, 0, stream>>>(dis, N);

  // ---- stack a ----
  k_gemm32<<<gemmBlocks, TPB, 0, stream>>>(x, W1a, t, N, 0);
  k_self_bias<<<nBlocksND, TPB, 0, stream>>>(t, dis, b1a, agg, N);
  k_edge_agg<<<nBlocksED, TPB, 0, stream>>>(t, dis, src, dst, agg, E);
  k_gemm32<<<gemmBlocks, TPB, 0, stream>>>(agg, W2a, t, N, 1);   // fused ReLU
  k_self_bias<<<nBlocksND, TPB, 0, stream>>>(t, dis, b2a, y1, N);
  k_edge_agg<<<nBlocksED, TPB, 0, stream>>>(t, dis, src, dst, y1, E);

  // ---- stack b ----
  k_gemm32<<<gemmBlocks, TPB, 0, stream>>>(x, W1b, t, N, 0);
  k_self_bias<<<nBlocksND, TPB, 0, stream>>>(t, dis, b1b, agg, N);
  k_edge_agg<<<nBlocksED, TPB, 0, stream>>>(t, dis, src, dst, agg, E);
  k_gemm32<<<gemmBlocks, TPB, 0, stream>>>(agg, W2b, t, N, 1);   // fused ReLU
  k_self_bias<<<nBlocksND, TPB, 0, stream>>>(t, dis, b2b, y2, N);
  k_edge_agg<<<nBlocksED, TPB, 0, stream>>>(t, dis, src, dst, y2, E);

  // ---- outputs ----
  k_dot<<<2, 32, 0, stream>>>(y1, y2, op1, op2, (float*)d_out);
}